// RandLANet_34385508172298
// MI455X (gfx1250) — compile-verified
//
#include <hip/hip_runtime.h>
#include <hip/hip_bf16.h>

// ---------------------------------------------------------------------------
// CDNA5 (gfx1250) WMMA fp32 fragments
// ---------------------------------------------------------------------------
typedef float v2f __attribute__((ext_vector_type(2)));
typedef float v8f __attribute__((ext_vector_type(8)));

// ---------------------------------------------------------------------------
// Fused linear kernel:  Y = act( bn( X @ W + bias ) + residual )
//   X : [M, CinP] row-major (CinP = Cin rounded up to 4; pad cols are zero)
//   Wt: [CoutP, CinP] = W^T zero-padded (CoutP = Cout rounded up to 16)
//   Y : [M, Cout] row-major
// M must be a multiple of 16 (true for every call site here).
// One wave computes one 16x16 tile via V_WMMA_F32_16X16X4_F32; the K loop is
// fully branchless: one global_load_b64 for A, one for B, then the WMMA.
// ---------------------------------------------------------------------------
__global__ __launch_bounds__(128)
void k_linear(const float* __restrict__ X, const float* __restrict__ Wt,
              const float* __restrict__ bias, const float* __restrict__ scale,
              const float* __restrict__ shift, const float* __restrict__ alpha,
              const float* __restrict__ residual, float* __restrict__ Y,
              int M, int CinP, int Cout, int actMode)
{
    const int wave = threadIdx.x >> 5;
    const int lane = threadIdx.x & 31;
    const int half = lane >> 4;          // 0: K pair {0,1}, 1: K pair {2,3}
    const int l16  = lane & 15;
    const int r0 = (blockIdx.x * 4 + wave) * 16;
    const int c0 = blockIdx.y * 16;
    if (r0 >= M) return;                 // wave-uniform exit

    // A: row (r0+l16), pairs at v2f index (k0>>1)+half
    const v2f* __restrict__ Arow = (const v2f*)(X  + (size_t)(r0 + l16) * CinP);
    // B: Wt row (c0+l16) holds W[:, col] contiguously in K
    const v2f* __restrict__ Brow = (const v2f*)(Wt + (size_t)(c0 + l16) * CinP);

    v8f acc = {0.f, 0.f, 0.f, 0.f, 0.f, 0.f, 0.f, 0.f};

    const int nIter = CinP >> 2;
    int kv = half;
    #pragma unroll 4
    for (int t = 0; t < nIter; ++t) {
        v2f a = Arow[kv];
        v2f b = Brow[kv];
        acc = __builtin_amdgcn_wmma_f32_16x16x4_f32(false, a, false, b,
                                                    (short)0, acc, false, false);
        kv += 2;
    }

    // ---- epilogue ----
    const int  col   = c0 + l16;
    const bool colOK = col < Cout;
    const int  colC  = colOK ? col : 0;
    const float b_v  = bias  ? bias[colC]  : 0.f;
    const float s_v  = scale ? scale[colC] : 1.f;
    const float sh_v = scale ? shift[colC] : 0.f;
    const float a_v  = alpha ? alpha[colC] : 0.f;

    #pragma unroll
    for (int j = 0; j < 8; ++j) {
        const int row = r0 + j + (half << 3);
        float y = acc[j] + b_v;
        y = y * s_v + sh_v;
        if (residual) y += residual[(size_t)row * Cout + colC];
        if (actMode == 1)      y = (y > 0.f) ? y : a_v * y;
        else if (actMode == 2) y = (y > 0.f) ? y : 0.2f * y;
        if (colOK) Y[(size_t)row * Cout + col] = y;
    }
}

// ---------------------------------------------------------------------------
// Weight packing: Wt[CoutP x CinP] = transpose(W[Cin x Cout]) zero-padded
// ---------------------------------------------------------------------------
__global__ void k_pack_w(const float* __restrict__ W, float* __restrict__ Wt,
                         int Cin, int Cout, int CinP, int CoutP)
{
    int gid = blockIdx.x * blockDim.x + threadIdx.x;
    int total = CoutP * CinP;
    if (gid >= total) return;
    int c = gid / CinP;
    int k = gid - c * CinP;
    Wt[gid] = (c < Cout && k < Cin) ? W[k * Cout + c] : 0.f;
}

// ---------------------------------------------------------------------------
// Elementwise / gather kernels
// ---------------------------------------------------------------------------

// features (B,3,N) -> X (B*N, 4) padded, scaled by 1/6
__global__ void k_transpose_in(const float* __restrict__ in, float* __restrict__ out,
                               int Bc, int N)
{
    int gid = blockIdx.x * blockDim.x + threadIdx.x;
    int total = Bc * N;
    if (gid >= total) return;
    int b = gid / N;
    int n = gid - b * N;
    const float* src = in + (size_t)b * 3 * N + n;
    float4 v;
    v.x = src[0]         * (1.0f / 6.0f);
    v.y = src[N]         * (1.0f / 6.0f);
    v.z = src[2 * (size_t)N] * (1.0f / 6.0f);
    v.w = 0.f;
    *(float4*)(out + (size_t)gid * 4) = v;
}

// 10-dim geometric feature per (b,n,k): [dist, rel(3), tile(3), nxyz(3)], stride 12
__global__ void k_fxyz(const float* __restrict__ xyz, const int* __restrict__ neigh,
                       float* __restrict__ out, int Bc, int N, int K)
{
    int gid = blockIdx.x * blockDim.x + threadIdx.x;
    int total = Bc * N * K;
    if (gid >= total) return;
    int b = gid / (N * K);
    int rem = gid - b * N * K;
    int n = rem / K;
    int idx = neigh[gid];
    const float* t  = xyz + ((size_t)b * N + n)   * 3;
    const float* nx = xyz + ((size_t)b * N + idx) * 3;
    float t0 = t[0], t1 = t[1], t2 = t[2];
    float x0 = nx[0], x1 = nx[1], x2 = nx[2];
    float rx = t0 - x0, ry = t1 - x1, rz = t2 - x2;
    float dist = sqrtf(rx * rx + ry * ry + rz * rz);
    float* o = out + (size_t)gid * 12;
    o[0] = dist; o[1] = rx; o[2] = ry; o[3]  = rz;
    o[4] = t0;   o[5] = t1; o[6] = t2; o[7]  = x0;
    o[8] = x1;   o[9] = x2; o[10] = 0.f; o[11] = 0.f;
}

// out[r, 0:C1] = src[b, neigh[r], 0:C1] ; out[r, C1:C1+C2] = direct[r, 0:C2]
__global__ void k_gather_concat(const float* __restrict__ src, const int* __restrict__ neigh,
                                const float* __restrict__ direct, float* __restrict__ out,
                                int Bc, int N, int K, int C1, int C2)
{
    int Ct = C1 + C2;
    long total = (long)Bc * N * K * Ct;
    long gid = (long)blockIdx.x * blockDim.x + threadIdx.x;
    if (gid >= total) return;
    int r = (int)(gid / Ct);
    int c = (int)(gid - (long)r * Ct);
    if (c < C1) {
        int b = r / (N * K);
        int idx = neigh[r];
        out[gid] = src[((size_t)b * N + idx) * C1 + c];
    } else {
        out[gid] = direct[(size_t)r * C2 + (c - C1)];
    }
}

// per (bn, c): softmax over K of S, then pooled = sum_k F * softmax(S)
__global__ void k_softmax_pool(const float* __restrict__ S, const float* __restrict__ F,
                               float* __restrict__ P, int BN, int K, int C)
{
    int gid = blockIdx.x * blockDim.x + threadIdx.x;
    int total = BN * C;
    if (gid >= total) return;
    int bn = gid / C;
    int c  = gid - bn * C;
    size_t base = (size_t)bn * K * C + c;
    float m = -3.0e38f;
    for (int k = 0; k < K; ++k) m = fmaxf(m, S[base + (size_t)k * C]);
    float sum = 0.f, acc = 0.f;
    for (int k = 0; k < K; ++k) {
        float e = __expf(S[base + (size_t)k * C] - m);
        sum += e;
        acc += F[base + (size_t)k * C] * e;
    }
    P[gid] = acc / sum;
}

// out[b,m,c] = max_k src[b, sub[b,m,k], c]
__global__ void k_gather_max(const float* __restrict__ src, const int* __restrict__ sub,
                             float* __restrict__ out, int Bc, int Nin, int M, int K, int C)
{
    int gid = blockIdx.x * blockDim.x + threadIdx.x;
    int total = Bc * M * C;
    if (gid >= total) return;
    int r = gid / C;              // b*M + m
    int c = gid - r * C;
    int b = r / M;
    float mx = -3.0e38f;
    for (int k = 0; k < K; ++k) {
        int idx = sub[(size_t)r * K + k];
        mx = fmaxf(mx, src[((size_t)b * Nin + idx) * C + c]);
    }
    out[gid] = mx;
}

// out[r, 0:C1] = encf[r, 0:C1] ; out[r, C1:] = fsrc[b, interp[r], 0:C2]
__global__ void k_concat_interp(const float* __restrict__ encf, const float* __restrict__ fsrc,
                                const int* __restrict__ interp, float* __restrict__ out,
                                int Bc, int M, int Nsub, int C1, int C2)
{
    int Ct = C1 + C2;
    long total = (long)Bc * M * Ct;
    long gid = (long)blockIdx.x * blockDim.x + threadIdx.x;
    if (gid >= total) return;
    int r = (int)(gid / Ct);
    int c = (int)(gid - (long)r * Ct);
    if (c < C1) {
        out[gid] = encf[(size_t)r * C1 + c];
    } else {
        int b = r / M;
        int idx = interp[r];
        out[gid] = fsrc[((size_t)b * Nsub + idx) * C2 + (c - C1)];
    }
}

// Y (B*N, C) -> out (B, C, N)
__global__ void k_transpose_out(const float* __restrict__ Yin, float* __restrict__ out,
                                int Bc, int N, int C)
{
    int gid = blockIdx.x * blockDim.x + threadIdx.x;
    int total = Bc * N * C;
    if (gid >= total) return;
    int b = gid / (N * C);
    int rem = gid - b * N * C;
    int n = rem / C;
    int c = rem - n * C;
    out[((size_t)b * C + c) * N + n] = Yin[gid];
}

// ---------------------------------------------------------------------------
// Host side
// ---------------------------------------------------------------------------
static inline int ceil_div(int a, int b) { return (a + b - 1) / b; }

struct Lin  { const float *w, *b, *s, *sh, *a; };
struct LinP { const float *wt, *b, *s, *sh, *a; int cinP, cout, coutP; };

static void run_linear(hipStream_t stream, const float* X, const LinP& L,
                       int M, float* Y, int actMode, const float* residual = nullptr)
{
    dim3 grid(ceil_div(M / 16, 4), L.coutP / 16);
    k_linear<<<grid, 128, 0, stream>>>(X, L.wt, L.b, L.s, L.sh, L.a,
                                       residual, Y, M, L.cinP, L.cout, actMode);
}

extern "C" void kernel_launch(void* const* d_in, const int* in_sizes, int n_in,
                              void* d_out, int out_size, void* d_ws, size_t ws_size,
                              hipStream_t stream)
{
    (void)in_sizes; (void)n_in; (void)out_size; (void)ws_size;

    constexpr int Bc = 2, K = 16;
    static const int Ns[5] = {45056, 11264, 2816, 704, 176};
    static const int Ds[4] = {16, 64, 128, 256};

    // ---- raw inputs (setup_inputs insertion order) ----
    const float* features = (const float*)d_in[0];
    const float* xyzP[4];  const int* neiP[4]; const int* subP[4]; const int* itpP[4];
    for (int i = 0; i < 4; ++i) {
        xyzP[i] = (const float*)d_in[1 + 4 * i];
        neiP[i] = (const int*)  d_in[2 + 4 * i];
        subP[i] = (const int*)  d_in[3 + 4 * i];
        itpP[i] = (const int*)  d_in[4 + 4 * i];
    }

    // ---- params: recursive insertion order of the params dict ----
    int pi = 17;
    auto take = [&](bool bias, bool bn, bool act) {
        Lin L{nullptr, nullptr, nullptr, nullptr, nullptr};
        L.w = (const float*)d_in[pi++];
        if (bias) L.b = (const float*)d_in[pi++];
        if (bn) { L.s = (const float*)d_in[pi++]; L.sh = (const float*)d_in[pi++]; }
        if (act) L.a = (const float*)d_in[pi++];
        return L;
    };
    Lin fc0r = take(false, true, true);
    struct EncR { Lin mlp1, bb1, att1fc, att1mlp, bb2, att2fc, att2mlp, mlp2, sc; } ER[4];
    for (int i = 0; i < 4; ++i) {
        ER[i].mlp1    = take(false, true,  true);
        ER[i].bb1     = take(false, true,  true);
        ER[i].att1fc  = take(false, false, false);
        ER[i].att1mlp = take(false, true,  true);
        ER[i].bb2     = take(false, true,  true);
        ER[i].att2fc  = take(false, false, false);
        ER[i].att2mlp = take(false, true,  true);
        ER[i].mlp2    = take(false, true,  false);
        ER[i].sc      = take(false, true,  false);
    }
    Lin decfr = take(false, true, true);
    Lin decr[4];
    for (int j = 0; j < 4; ++j) decr[j] = take(false, true, true);
    Lin fc1r = take(false, true, true);
    Lin fc2r = take(false, true, true);
    Lin fc3r = take(true,  false, false);

    // ---- workspace arena (floats) ----
    float* ws = (float*)d_ws;
    size_t off = 0;
    auto alloc = [&](size_t n) { float* p = ws + off; off += n; return p; };

    // pack weights: transposed + padded (CinP x4, CoutP x16)
    auto pack = [&](const Lin& L, int cin, int cout) {
        int cinP  = (cin  + 3)  & ~3;
        int coutP = (cout + 15) & ~15;
        float* wt = alloc((size_t)cinP * coutP);
        int tot = cinP * coutP;
        k_pack_w<<<ceil_div(tot, 256), 256, 0, stream>>>(L.w, wt, cin, cout, cinP, coutP);
        LinP p{wt, L.b, L.s, L.sh, L.a, cinP, cout, coutP};
        return p;
    };

    LinP fc0 = pack(fc0r, 3, 8);
    struct Enc { LinP mlp1, bb1, att1fc, att1mlp, bb2, att2fc, att2mlp, mlp2, sc; } E[4];
    {
        int din = 8;
        for (int i = 0; i < 4; ++i) {
            int d = Ds[i], h = d / 2;
            E[i].mlp1    = pack(ER[i].mlp1,    din, h);
            E[i].bb1     = pack(ER[i].bb1,     10,  h);
            E[i].att1fc  = pack(ER[i].att1fc,  d,   d);
            E[i].att1mlp = pack(ER[i].att1mlp, d,   h);
            E[i].bb2     = pack(ER[i].bb2,     h,   h);
            E[i].att2fc  = pack(ER[i].att2fc,  d,   d);
            E[i].att2mlp = pack(ER[i].att2mlp, d,   d);
            E[i].mlp2    = pack(ER[i].mlp2,    d,   2 * d);
            E[i].sc      = pack(ER[i].sc,      din, 2 * d);
            din = 2 * d;
        }
    }
    LinP decf = pack(decfr, 512, 512);
    LinP dec[4] = { pack(decr[0], 768, 256), pack(decr[1], 384, 128),
                    pack(decr[2], 160, 32),  pack(decr[3], 64, 32) };
    LinP fc1 = pack(fc1r, 32, 64);
    LinP fc2 = pack(fc2r, 64, 32);
    LinP fc3 = pack(fc3r, 32, 19);

    // activation buffers
    float* X0     = alloc((size_t)Bc * Ns[0] * 4);   // padded input (stride 4)
    float* F8     = alloc((size_t)Bc * Ns[0] * 8);   // fc0 out
    float* bufA   = alloc(23068672);                 // K-expanded concat / decoder cat
    float* bufB   = alloc(23068672);                 // attention scores / decoder out
    float* bufC   = alloc(17301504);                 // f_xyz12 / f_xyz2 (stride-12 max)
    float* bufD   = alloc(11534336);                 // f_xyz (bb_mlp1 out)
    float* fpc    = alloc(720896);                   // mlp1 out (P x d/2)
    float* pool   = alloc(1441792);                  // attention pooled (P x d)
    float* fagg   = alloc(720896);                   // att1_mlp out (P x d/2)
    float* fbb    = alloc(1441792);                  // att2_mlp out (P x d)
    float* fpc2   = alloc(2883584);                  // mlp2 out (P x 2d)
    float* enc0   = alloc(2883584);                  // persistent f_enc layer0 (B*N0 x 32)
    float* fenc_t = alloc(2883584);                  // f_enc temp for layers 1..3
    float* f0     = alloc(720896);                   // B*N1 x 32
    float* f1     = alloc(720896);                   // B*N2 x 128
    float* f2     = alloc(360448);                   // B*N3 x 256
    float* f3     = alloc(180224);                   // B*N4 x 512
    float* g      = alloc(180224);                   // dec_first out (B*N4 x 512)
    float* fouts[4] = {f0, f1, f2, f3};

    const int TB = 256;

    // ---- stem ----
    {
        int tot = Bc * Ns[0];
        k_transpose_in<<<ceil_div(tot, TB), TB, 0, stream>>>(features, X0, Bc, Ns[0]);
        run_linear(stream, X0, fc0, Bc * Ns[0], F8, 1);
    }

    // ---- encoder ----
    const float* f_in = F8;
    for (int i = 0; i < 4; ++i) {
        const int d = Ds[i], h = d / 2;
        const int N = Ns[i], P = Bc * N, R = P * K;

        run_linear(stream, f_in, E[i].mlp1, P, fpc, 1);
        k_fxyz<<<ceil_div(R, TB), TB, 0, stream>>>(xyzP[i], neiP[i], bufC, Bc, N, K);
        run_linear(stream, bufC, E[i].bb1, R, bufD, 1);
        { int tot = R * d;
          k_gather_concat<<<ceil_div(tot, TB), TB, 0, stream>>>(fpc, neiP[i], bufD, bufA, Bc, N, K, h, h); }
        run_linear(stream, bufA, E[i].att1fc, R, bufB, 0);
        { int tot = P * d;
          k_softmax_pool<<<ceil_div(tot, TB), TB, 0, stream>>>(bufB, bufA, pool, P, K, d); }
        run_linear(stream, pool, E[i].att1mlp, P, fagg, 1);
        run_linear(stream, bufD, E[i].bb2, R, bufC, 1);
        { int tot = R * d;
          k_gather_concat<<<ceil_div(tot, TB), TB, 0, stream>>>(fagg, neiP[i], bufC, bufA, Bc, N, K, h, h); }
        run_linear(stream, bufA, E[i].att2fc, R, bufB, 0);
        { int tot = P * d;
          k_softmax_pool<<<ceil_div(tot, TB), TB, 0, stream>>>(bufB, bufA, pool, P, K, d); }
        run_linear(stream, pool, E[i].att2mlp, P, fbb, 1);
        run_linear(stream, fbb, E[i].mlp2, P, fpc2, 0);
        float* fenc = (i == 0) ? enc0 : fenc_t;
        run_linear(stream, f_in, E[i].sc, P, fenc, 2, fpc2);
        { int M2 = Ns[i + 1]; int tot = Bc * M2 * 2 * d;
          k_gather_max<<<ceil_div(tot, TB), TB, 0, stream>>>(fenc, subP[i], fouts[i], Bc, N, M2, K, 2 * d); }

        f_in = fouts[i];
    }

    // ---- decoder ----
    run_linear(stream, f3, decf, Bc * Ns[4], g, 1);

    { int M = Ns[3]; int tot = Bc * M * 768;   // cat(f2[256], interp3(g)[512]) -> dec0 -> 256
      k_concat_interp<<<ceil_div(tot, TB), TB, 0, stream>>>(f2, g, itpP[3], bufA, Bc, M, Ns[4], 256, 512);
      run_linear(stream, bufA, dec[0], Bc * M, bufB, 1); }
    { int M = Ns[2]; int tot = Bc * M * 384;   // cat(f1[128], interp2(d0)[256]) -> dec1 -> 128
      k_concat_interp<<<ceil_div(tot, TB), TB, 0, stream>>>(f1, bufB, itpP[2], bufA, Bc, M, Ns[3], 128, 256);
      run_linear(stream, bufA, dec[1], Bc * M, bufB, 1); }
    { int M = Ns[1]; int tot = Bc * M * 160;   // cat(f0[32], interp1(d1)[128]) -> dec2 -> 32
      k_concat_interp<<<ceil_div(tot, TB), TB, 0, stream>>>(f0, bufB, itpP[1], bufA, Bc, M, Ns[2], 32, 128);
      run_linear(stream, bufA, dec[2], Bc * M, bufB, 1); }
    { int M = Ns[0]; int tot = Bc * M * 64;    // cat(enc0[32], interp0(d2)[32]) -> dec3 -> 32
      k_concat_interp<<<ceil_div(tot, TB), TB, 0, stream>>>(enc0, bufB, itpP[0], bufA, Bc, M, Ns[1], 32, 32);
      run_linear(stream, bufA, dec[3], Bc * M, bufB, 1); }

    // ---- head ----
    run_linear(stream, bufB, fc1, Bc * Ns[0], bufA, 1);
    run_linear(stream, bufA, fc2, Bc * Ns[0], bufB, 1);
    run_linear(stream, bufB, fc3, Bc * Ns[0], bufA, 0);
    { int tot = Bc * Ns[0] * 19;
      k_transpose_out<<<ceil_div(tot, TB), TB, 0, stream>>>(bufA, (float*)d_out, Bc, Ns[0], 19); }
}